// GCNDecoder_3719441678954
// MI455X (gfx1250) — compile-verified
//
#include <hip/hip_runtime.h>
#include <math.h>

#define B_   128
#define NN_  400
#define FEAT_ 256
#define DEG_  64
#define NTOT (B_*NN_)        // 51200
#define ETOT (NTOT*DEG_)     // 3276800

typedef __attribute__((ext_vector_type(2))) float v2f;
typedef __attribute__((ext_vector_type(8))) float v8f;

__device__ __forceinline__ float lrelu(float v) { return v >= 0.f ? v : 0.01f * v; }

__global__ void k_zero(float* __restrict__ p, int n) {
  int i = blockIdx.x * blockDim.x + threadIdx.x;
  int stride = gridDim.x * blockDim.x;
  for (; i < n; i += stride) p[i] = 0.f;
}

// h1[b*NN+n] = leaky_relu(dot(x[b,:], fc1_w[n,:]) + fc1_b[n])
__global__ void k_fc1(const float* __restrict__ x, const float* __restrict__ w,
                      const float* __restrict__ bias, float* __restrict__ h1) {
  __shared__ float xs[FEAT_];
  int b = blockIdx.x;
  for (int f = threadIdx.x; f < FEAT_; f += blockDim.x) xs[f] = x[b * FEAT_ + f];
  __syncthreads();
  int n = threadIdx.x;
  if (n < NN_) {
    const float* wr = w + n * FEAT_;      // row is 1KB-aligned -> float4 ok
    float acc = bias[n];
    #pragma unroll 8
    for (int f = 0; f < FEAT_; f += 4) {
      float4 wv = *(const float4*)(wr + f);
      acc += xs[f] * wv.x + xs[f + 1] * wv.y + xs[f + 2] * wv.z + xs[f + 3] * wv.w;
    }
    h1[b * NN_ + n] = lrelu(acc);
  }
}

// conv1 edge stage (C=1): agg1[dst] += h1[src]*ew ; cnt[dst] += 1
__global__ void k_edge1(const int* __restrict__ src, const int* __restrict__ dst,
                        const float* __restrict__ ea, const float* __restrict__ h1,
                        float* __restrict__ agg1, float* __restrict__ cnt) {
  int e = blockIdx.x * blockDim.x + threadIdx.x;
  if (e < ETOT) {
    int s = src[e], d = dst[e];
    atomicAdd(&agg1[d], h1[s] * ea[e]);
    atomicAdd(&cnt[d], 1.0f);
  }
}

// conv1 node stage: h2[n,c] = lrelu(mean1[n]*W_rel1[c] + b_rel1[c] + h1[n]*W_root1[c])
__global__ void k_node1(const float* __restrict__ h1, const float* __restrict__ agg1,
                        const float* __restrict__ cnt, const float* __restrict__ Wrel1,
                        const float* __restrict__ brel1, const float* __restrict__ Wroot1,
                        float* __restrict__ h2) {
  int n = blockIdx.x * blockDim.x + threadIdx.x;
  if (n < NTOT) {
    float inv = 1.f / fmaxf(cnt[n], 1.f);
    float a = agg1[n] * inv;
    float hv = h1[n];
    #pragma unroll
    for (int c = 0; c < 8; ++c)
      h2[n * 8 + c] = lrelu(a * Wrel1[c] + brel1[c] + hv * Wroot1[c]);
  }
}

// conv2 edge stage: one thread per (edge, channel) for coalesced gather + atomics
__global__ void k_edge2(const int* __restrict__ src, const int* __restrict__ dst,
                        const float* __restrict__ ea, const float* __restrict__ h2,
                        float* __restrict__ agg2) {
  int t = blockIdx.x * blockDim.x + threadIdx.x;
  if (t < ETOT * 8) {
    int e = t >> 3, c = t & 7;
    int s = src[e], d = dst[e];
    atomicAdd(&agg2[d * 8 + c], h2[s * 8 + c] * ea[e]);
  }
}

// conv2 node stage fused with P build: P[n,0:8]=agg2/cnt, P[n,8:16]=h2
__global__ void k_buildP(const float* __restrict__ agg2, const float* __restrict__ h2,
                         const float* __restrict__ cnt, float* __restrict__ P) {
  int t = blockIdx.x * blockDim.x + threadIdx.x;
  if (t < NTOT * 8) {
    int n = t >> 3, c = t & 7;
    float inv = 1.f / fmaxf(cnt[n], 1.f);
    P[n * 16 + c]     = agg2[n * 8 + c] * inv;
    P[n * 16 + 8 + c] = h2[n * 8 + c];
  }
}

// Wc[j,0:8]=W_rel2[j,:], Wc[j,8:16]=W_root2[j,:]
__global__ void k_wcat(const float* __restrict__ Wrel2, const float* __restrict__ Wroot2,
                       float* __restrict__ Wc) {
  int t = blockIdx.x * blockDim.x + threadIdx.x;
  if (t < NN_ * 8) {
    int j = t >> 3, c = t & 7;
    Wc[j * 16 + c]     = Wrel2[j * 8 + c];
    Wc[j * 16 + 8 + c] = Wroot2[j * 8 + c];
  }
}

// Final fused stage: per (batch, 16x16 tile), D = P_i·W_jT + W_i·P_jT (dual WMMA:
// second product equals the transposed tile elementwise -> symmetric sum w/o shuffles),
// then out = sigmoid(0.5*(D + b[i] + b[j])).
// V_WMMA_F32_16X16X4_F32 operand layout (ISA 7.12.2): lane%16 = row, K = vgpr + 2*(lane/16).
__global__ void k_out(const float* __restrict__ P, const float* __restrict__ Wc,
                      const float* __restrict__ brel2, float* __restrict__ out) {
  int t = blockIdx.x * (blockDim.x >> 5) + (threadIdx.x >> 5);  // global wave id
  int b = t / 625;
  int rem = t - b * 625;
  int i0 = (rem / 25) * 16;
  int j0 = (rem % 25) * 16;
  int lane = threadIdx.x & 31;
  int half = lane >> 4;
  int lm = lane & 15;
  const float* Pb = P + b * (NN_ * 16);

  v8f acc = {0.f, 0.f, 0.f, 0.f, 0.f, 0.f, 0.f, 0.f};
  #pragma unroll
  for (int k4 = 0; k4 < 16; k4 += 4) {
    int ko = k4 + 2 * half;   // float2 -> (K=ko in vgpr0, K=ko+1 in vgpr1)
    v2f a1 = *(const v2f*)(Pb + (i0 + lm) * 16 + ko);  // A = P rows i-tile
    v2f b1 = *(const v2f*)(Wc + (j0 + lm) * 16 + ko);  // B = Wc rows j-tile (as cols)
    v2f a2 = *(const v2f*)(Wc + (i0 + lm) * 16 + ko);  // A = Wc rows i-tile
    v2f b2 = *(const v2f*)(Pb + (j0 + lm) * 16 + ko);  // B = P rows j-tile (as cols)
    acc = __builtin_amdgcn_wmma_f32_16x16x4_f32(false, a1, false, b1, (short)0, acc, false, false);
    acc = __builtin_amdgcn_wmma_f32_16x16x4_f32(false, a2, false, b2, (short)0, acc, false, false);
  }

  float bj = brel2[j0 + lm];
  float* ob = out + (size_t)b * NN_ * NN_;
  #pragma unroll
  for (int v = 0; v < 8; ++v) {
    int M = v + 8 * half;                     // C/D layout: vgpr v, lane half -> row
    float z = 0.5f * (acc[v] + brel2[i0 + M] + bj);
    ob[(i0 + M) * NN_ + (j0 + lm)] = 1.f / (1.f + expf(-z));
  }
}

extern "C" void kernel_launch(void* const* d_in, const int* in_sizes, int n_in,
                              void* d_out, int out_size, void* d_ws, size_t ws_size,
                              hipStream_t stream) {
  const float* x      = (const float*)d_in[0];
  const int*   ei     = (const int*)  d_in[1];
  const float* ea     = (const float*)d_in[2];
  const float* fc1w   = (const float*)d_in[3];
  const float* fc1b   = (const float*)d_in[4];
  const float* Wrel1  = (const float*)d_in[5];
  const float* brel1  = (const float*)d_in[6];
  const float* Wroot1 = (const float*)d_in[7];
  const float* Wrel2  = (const float*)d_in[8];
  const float* brel2  = (const float*)d_in[9];
  const float* Wroot2 = (const float*)d_in[10];
  float* out = (float*)d_out;

  float* ws   = (float*)d_ws;
  float* h1   = ws;                    // NTOT
  float* cnt  = h1 + NTOT;             // NTOT
  float* agg1 = cnt + NTOT;            // NTOT
  float* h2   = agg1 + NTOT;           // NTOT*8
  float* agg2 = h2 + (size_t)NTOT * 8; // NTOT*8
  float* P    = agg2 + (size_t)NTOT * 8; // NTOT*16
  float* Wc   = P + (size_t)NTOT * 16;   // NN*16  (total ~7.2 MB)

  const int* srcI = ei;
  const int* dstI = ei + ETOT;

  k_zero<<<512, 256, 0, stream>>>(cnt, NTOT * 2);        // cnt + agg1 (contiguous)
  k_zero<<<1024, 256, 0, stream>>>(agg2, NTOT * 8);

  k_fc1<<<B_, 512, 0, stream>>>(x, fc1w, fc1b, h1);

  k_edge1<<<(ETOT + 255) / 256, 256, 0, stream>>>(srcI, dstI, ea, h1, agg1, cnt);
  k_node1<<<(NTOT + 255) / 256, 256, 0, stream>>>(h1, agg1, cnt, Wrel1, brel1, Wroot1, h2);

  k_edge2<<<(ETOT * 8 + 255) / 256, 256, 0, stream>>>(srcI, dstI, ea, h2, agg2);
  k_buildP<<<(NTOT * 8 + 255) / 256, 256, 0, stream>>>(agg2, h2, cnt, P);
  k_wcat<<<(NN_ * 8 + 255) / 256, 256, 0, stream>>>(Wrel2, Wroot2, Wc);

  // 128 batches * 625 tiles = 80000 waves, 8 waves (256 thr) per block -> 10000 blocks
  k_out<<<(B_ * 625) / 8, 256, 0, stream>>>(P, Wc, brel2, out);
}